// NodeEmbedding_62766652064051
// MI455X (gfx1250) — compile-verified
//
#include <hip/hip_runtime.h>

#define DIM 128
#define BN_EPS 1e-5f
#define L2_EPS 1e-12f

typedef __attribute__((ext_vector_type(2))) float v2f;
typedef __attribute__((ext_vector_type(8))) float v8f;

// ---------------------------------------------------------------- init
__global__ __launch_bounds__(256) void k_init(float* __restrict__ deg,
                                              float* __restrict__ fsum,
                                              float* __restrict__ fsq, int N) {
  int i = blockIdx.x * blockDim.x + threadIdx.x;
  if (i < N) deg[i] = 1.0f;          // self-loop contributes 1 to every degree
  if (i < DIM) { fsum[i] = 0.0f; fsq[i] = 0.0f; }
}

// ---------------------------------------------------------------- degrees
__global__ __launch_bounds__(256) void k_degree(const int* __restrict__ ecol,
                                                float* __restrict__ deg, int E) {
  int e = blockIdx.x * blockDim.x + threadIdx.x;
  if (e < E) unsafeAtomicAdd(&deg[ecol[e]], 1.0f);
}

__global__ __launch_bounds__(256) void k_rsqrt(const float* __restrict__ deg,
                                               float* __restrict__ dis, int N) {
  int i = blockIdx.x * blockDim.x + threadIdx.x;
  if (i < N) dis[i] = rsqrtf(deg[i]);   // deg >= 1 always (self loops)
}

// ---------------------------------------------------------------- GEMM h = x @ W
// One block handles 16 rows of x; the 16x128 row-slab is staged in LDS
// (stride padded to 132 floats to spread the 64 LDS banks).
// 8 waves per block; wave w computes the 16x16 output tile at cols [16w,16w+16).
// K=128 consumed as 32 x V_WMMA_F32_16X16X4_F32.
__global__ __launch_bounds__(256) void k_gemm_wmma(const float* __restrict__ x,
                                                   const float* __restrict__ W,
                                                   float* __restrict__ h, int N) {
  __shared__ float lx[16 * 132];
  const int tid  = threadIdx.x;
  const int wave = tid >> 5;       // 0..7 -> 16-col tile
  const int lane = tid & 31;
  const int m0   = blockIdx.x * 16;

  // cooperative load: 256 threads x 8 floats = 16 rows x 128 cols
  {
    int r    = tid >> 4;           // 0..15
    int cseg = (tid & 15) * 8;     // 0..120
    int gr   = m0 + r; if (gr >= N) gr = N - 1;
    const float* src = x + (size_t)gr * DIM + cseg;
    float*       dst = lx + r * 132 + cseg;
#pragma unroll
    for (int j = 0; j < 8; ++j) dst[j] = src[j];
  }
  __syncthreads();

  const int half = lane >> 4;      // 0: lanes 0-15, 1: lanes 16-31
  const int lm   = lane & 15;
  const int n0   = wave * 16;

  v8f c = {};

#if __has_builtin(__builtin_amdgcn_wmma_f32_16x16x4_f32)
#pragma unroll 8
  for (int k0 = 0; k0 < DIM; k0 += 4) {
    const int ka = k0 + 2 * half;
    v2f a, b;
    // A 16x4 fragment: lane (half,lm) holds A[lm][ka], A[lm][ka+1]
    a.x = lx[lm * 132 + ka];
    a.y = lx[lm * 132 + ka + 1];
    // B 4x16 fragment: lane (half,lm) holds B[ka][lm], B[ka+1][lm]
    b.x = W[(size_t)ka * DIM + n0 + lm];
    b.y = W[(size_t)(ka + 1) * DIM + n0 + lm];
    c = __builtin_amdgcn_wmma_f32_16x16x4_f32(false, a, false, b,
                                              (short)0, c, false, false);
  }
#else
  // scalar fallback producing the identical C/D register layout
  for (int k = 0; k < DIM; ++k) {
#pragma unroll
    for (int r = 0; r < 8; ++r)
      c[r] = fmaf(lx[(r + 8 * half) * 132 + k], W[(size_t)k * DIM + n0 + lm], c[r]);
  }
#endif

  // D layout: VGPR r -> row M = r + 8*half, col N = lm.
  // Single tile-level guard keeps the hot path free of per-row exec masking
  // (N % 16 == 0 in practice, so the fast path always taken).
  float* hp = h + (size_t)(m0 + 8 * half) * DIM + n0 + lm;
  if (m0 + 16 <= N) {
#pragma unroll
    for (int r = 0; r < 8; ++r) hp[(size_t)r * DIM] = c[r];
  } else {
#pragma unroll
    for (int r = 0; r < 8; ++r)
      if (m0 + r + 8 * half < N) hp[(size_t)r * DIM] = c[r];
  }
}

// ---------------------------------------------------------------- self-loop seed
// out[i,:] = h[i,:] * dis[i]^2   (initializes every element of d_out)
__global__ __launch_bounds__(256) void k_selfloop(const float* __restrict__ h,
                                                  const float* __restrict__ dis,
                                                  float* __restrict__ out,
                                                  long long total) {
  long long i = (long long)blockIdx.x * blockDim.x + threadIdx.x;
  if (i >= total) return;
  int node = (int)(i >> 7);
  float s = dis[node];
  out[i] = h[i] * s * s;
}

// ---------------------------------------------------------------- edge scatter
// thread = (edge, feature); gathers h[row], scales by dis[row]*dis[col],
// scatter-adds into out[col]. Consecutive threads share an edge -> coalesced;
// h and out are both L2-resident (51 MB each vs 192 MB L2).
__global__ __launch_bounds__(256) void k_scatter(const float* __restrict__ h,
                                                 const float* __restrict__ dis,
                                                 const int* __restrict__ erow,
                                                 const int* __restrict__ ecol,
                                                 float* __restrict__ out,
                                                 long long total) {
  long long i = (long long)blockIdx.x * blockDim.x + threadIdx.x;
  if (i >= total) return;
  int e = (int)(i >> 7);
  int d = (int)(i & 127);
  int r = erow[e], c = ecol[e];
  float v = h[(size_t)r * DIM + d] * (dis[r] * dis[c]);
  unsafeAtomicAdd(&out[(size_t)c * DIM + d], v);
}

// ---------------------------------------------------------------- bias+ReLU+stats
// grid stride is a multiple of 128, so each thread owns a fixed feature d.
__global__ __launch_bounds__(256) void k_bias_relu_stats(float* __restrict__ out,
                                                         const float* __restrict__ bias,
                                                         float* __restrict__ fsum,
                                                         float* __restrict__ fsq,
                                                         long long total) {
  __shared__ float s0[256], s1[256];
  const int tid = threadIdx.x;
  const int d = tid & 127;
  const float b = bias[d];
  float ls = 0.0f, lq = 0.0f;
  const long long stride = (long long)gridDim.x * blockDim.x;
  for (long long i = (long long)blockIdx.x * blockDim.x + tid; i < total; i += stride) {
    float v = out[i] + b;
    v = fmaxf(v, 0.0f);
    out[i] = v;
    ls += v;
    lq += v * v;
  }
  s0[tid] = ls; s1[tid] = lq;
  __syncthreads();
  if (tid < 128) {
    unsafeAtomicAdd(&fsum[d], s0[tid] + s0[tid + 128]);
    unsafeAtomicAdd(&fsq[d],  s1[tid] + s1[tid + 128]);
  }
}

// ---------------------------------------------------------------- BN + L2 norm
// wave32 per row; each lane owns 4 features (float4); shuffle-reduce the
// squared norm across the 32 lanes.
__global__ __launch_bounds__(256) void k_bn_l2(float* __restrict__ out,
                                               const float* __restrict__ fsum,
                                               const float* __restrict__ fsq,
                                               const float* __restrict__ gamma,
                                               const float* __restrict__ beta,
                                               int N) {
  const int wave = threadIdx.x >> 5;
  const int lane = threadIdx.x & 31;
  const int node = blockIdx.x * 8 + wave;
  if (node >= N) return;
  const float invN = 1.0f / (float)N;
  float* rowp = out + (size_t)node * DIM + lane * 4;
  const float4 v = *(const float4*)rowp;
  const float vv[4] = {v.x, v.y, v.z, v.w};
  float ys[4];
  float ss = 0.0f;
#pragma unroll
  for (int j = 0; j < 4; ++j) {
    int dd = lane * 4 + j;
    float mean  = fsum[dd] * invN;
    float var   = fsq[dd] * invN - mean * mean;
    float scale = gamma[dd] * rsqrtf(var + BN_EPS);
    float y = (vv[j] - mean) * scale + beta[dd];
    ys[j] = y;
    ss += y * y;
  }
#pragma unroll
  for (int off = 16; off > 0; off >>= 1) ss += __shfl_xor(ss, off, 32);
  const float inv = 1.0f / fmaxf(sqrtf(ss), L2_EPS);
  float4 o;
  o.x = ys[0] * inv; o.y = ys[1] * inv; o.z = ys[2] * inv; o.w = ys[3] * inv;
  *(float4*)rowp = o;
}

// ---------------------------------------------------------------- launch
extern "C" void kernel_launch(void* const* d_in, const int* in_sizes, int n_in,
                              void* d_out, int out_size, void* d_ws, size_t ws_size,
                              hipStream_t stream) {
  const float* x     = (const float*)d_in[0];
  const float* W     = (const float*)d_in[1];
  const float* bias  = (const float*)d_in[2];
  const float* gamma = (const float*)d_in[3];
  const float* beta  = (const float*)d_in[4];
  const int*   ei    = (const int*)d_in[5];

  const int N = in_sizes[0] / DIM;
  const int E = in_sizes[5] / 2;
  const int* erow = ei;       // sources
  const int* ecol = ei + E;   // targets

  float* out = (float*)d_out;

  // workspace layout: h[N*128] | deg[N] | dis[N] | fsum[128] | fsq[128]
  float* h    = (float*)d_ws;
  float* deg  = h + (size_t)N * DIM;
  float* dis  = deg + N;
  float* fsum = dis + N;
  float* fsq  = fsum + DIM;

  const long long totalND = (long long)N * DIM;
  const long long totalE  = (long long)E * DIM;

  k_init  <<<(N + 255) / 256, 256, 0, stream>>>(deg, fsum, fsq, N);
  k_degree<<<(E + 255) / 256, 256, 0, stream>>>(ecol, deg, E);
  k_rsqrt <<<(N + 255) / 256, 256, 0, stream>>>(deg, dis, N);
  k_gemm_wmma<<<(N + 15) / 16, 256, 0, stream>>>(x, W, h, N);
  k_selfloop<<<(unsigned)((totalND + 255) / 256), 256, 0, stream>>>(h, dis, out, totalND);
  k_scatter <<<(unsigned)((totalE + 255) / 256), 256, 0, stream>>>(h, dis, erow, ecol, out, totalE);
  k_bias_relu_stats<<<512, 256, 0, stream>>>(out, bias, fsum, fsq, totalND);
  k_bn_l2<<<(N + 7) / 8, 256, 0, stream>>>(out, fsum, fsq, gamma, beta, N);
}